// GCNLayer_12317966205308
// MI455X (gfx1250) — compile-verified
//
#include <hip/hip_runtime.h>
#include <hip/hip_bf16.h>
#include <stdint.h>

#define D_FEAT 64

typedef __attribute__((ext_vector_type(2))) float v2f;
typedef __attribute__((ext_vector_type(8))) float v8f;

// ---------------------------------------------------------------------------
// Kernel 1: xw = x @ W  using V_WMMA_F32_16X16X4_F32 (full f32 precision).
// Block = 128 threads = 4 wave32s. Each block computes a 16x64 tile of xw;
// wave w owns output columns [16w, 16w+16). K=64 -> 16 WMMA steps of K=4.
// W (64x64) and the 16x64 x-tile are staged in LDS with padded strides to
// avoid the 256B-stride bank aliasing (64 banks x 4B).
// ---------------------------------------------------------------------------
__global__ __launch_bounds__(128) void gcn_gemm_wmma(
    const float* __restrict__ x, const float* __restrict__ W,
    float* __restrict__ xw)
{
    __shared__ __align__(16) float sX[16 * 66];   // x tile, stride 66
    __shared__ __align__(16) float sW[64 * 80];   // W,      stride 80

    const int tid  = threadIdx.x;
    const int row0 = blockIdx.x * 16;

    // Stage x tile (16x64) -> LDS
    for (int i = tid; i < (16 * 64) / 4; i += 128) {
        const int r = (i * 4) / 64;
        const int c = (i * 4) % 64;
        const float4 v = *(const float4*)(x + (size_t)(row0 + r) * D_FEAT + c);
        float* p = &sX[r * 66 + c];
        p[0] = v.x; p[1] = v.y; p[2] = v.z; p[3] = v.w;
    }
    // Stage W (64x64) -> LDS
    for (int i = tid; i < (64 * 64) / 4; i += 128) {
        const int r = (i * 4) / 64;
        const int c = (i * 4) % 64;
        const float4 v = *(const float4*)(W + i * 4);
        float* p = &sW[r * 80 + c];
        p[0] = v.x; p[1] = v.y; p[2] = v.z; p[3] = v.w;
    }
    __syncthreads();

    const int wave = tid >> 5;
    const int lane = tid & 31;
    const int half = lane >> 4;   // K-half select for A/B fragments
    const int r    = lane & 15;   // M index (A) / N index (B)
    const int cb   = wave * 16;   // output-column base of this wave

    v8f acc = {};
#pragma unroll
    for (int kk = 0; kk < 64; kk += 4) {
        const int ka = kk + 2 * half;
        // A frag (16x4 f32): VGPR v holds x[row0+r][ka+v]
        const float2 av = *(const float2*)(&sX[r * 66 + ka]);
        v2f a; a.x = av.x; a.y = av.y;
        // B frag (4x16 f32): VGPR v holds W[ka+v][cb+r]
        v2f b;
        b.x = sW[(ka + 0) * 80 + cb + r];
        b.y = sW[(ka + 1) * 80 + cb + r];
        // (neg_a, A, neg_b, B, c_mod, C, reuse_a, reuse_b)
        acc = __builtin_amdgcn_wmma_f32_16x16x4_f32(
            false, a, false, b, (short)0, acc, false, false);
    }

    // D layout (16x16 f32 C/D): acc[v] -> row (v + 8*half), col r
#pragma unroll
    for (int v = 0; v < 8; ++v) {
        xw[(size_t)(row0 + v + 8 * half) * D_FEAT + cb + r] = acc[v];
    }
}

// ---------------------------------------------------------------------------
// Sparse part: degree -> rsqrt -> bias/self-loop init -> edge scatter -> ReLU
// ---------------------------------------------------------------------------
__global__ void k_deg_init(float* __restrict__ deg, int n) {
    int i = blockIdx.x * blockDim.x + threadIdx.x;
    if (i < n) deg[i] = 1.0f;   // self-loop contributes 1 to every target
}

__global__ void k_deg_count(const long long* __restrict__ dst,
                            float* __restrict__ deg, int e) {
    int i = blockIdx.x * blockDim.x + threadIdx.x;
    if (i < e) atomicAdd(&deg[(int)dst[i]], 1.0f);
}

__global__ void k_dis(float* __restrict__ deg, int n) {
    int i = blockIdx.x * blockDim.x + threadIdx.x;
    if (i < n) deg[i] = rsqrtf(deg[i]);   // deg >= 1 always (self-loops)
}

// out[i][d] = b[d] + dis[i]^2 * xw[i][d]   (self-loop message folded in;
// full overwrite of d_out so the 0xAA poison never survives)
__global__ void k_out_init(const float* __restrict__ xw,
                           const float* __restrict__ dis,
                           const float* __restrict__ b,
                           float* __restrict__ out, int total) {
    int t = blockIdx.x * blockDim.x + threadIdx.x;
    if (t < total) {
        const int i = t >> 6;
        const int d = t & 63;
        const float s = dis[i];
        out[t] = b[d] + s * s * xw[t];
    }
}

// 16 threads per edge; float4 gather of xw[src], 4 f32 atomics into out[dst]
__global__ void k_scatter(const long long* __restrict__ src,
                          const long long* __restrict__ dst,
                          const float* __restrict__ xw,
                          const float* __restrict__ dis,
                          float* __restrict__ out, int e) {
    int t = blockIdx.x * blockDim.x + threadIdx.x;
    if (t < e * 16) {
        const int ed = t >> 4;
        const int q  = t & 15;
        const int s  = (int)src[ed];
        const int d  = (int)dst[ed];
        const float c = dis[s] * dis[d];
        const float4 v = *((const float4*)(xw + (size_t)s * D_FEAT) + q);
        float* o = out + (size_t)d * D_FEAT + q * 4;
        atomicAdd(o + 0, c * v.x);
        atomicAdd(o + 1, c * v.y);
        atomicAdd(o + 2, c * v.z);
        atomicAdd(o + 3, c * v.w);
    }
}

__global__ void k_relu(float* __restrict__ out, int total) {
    int t = blockIdx.x * blockDim.x + threadIdx.x;
    if (t < total) out[t] = fmaxf(out[t], 0.0f);
}

// ---------------------------------------------------------------------------
extern "C" void kernel_launch(void* const* d_in, const int* in_sizes, int n_in,
                              void* d_out, int out_size, void* d_ws, size_t ws_size,
                              hipStream_t stream) {
    const float*     x  = (const float*)d_in[0];        // [N, 64]
    const long long* ei = (const long long*)d_in[1];    // [2, E] int64
    const float*     W  = (const float*)d_in[2];        // [64, 64]
    const float*     b  = (const float*)d_in[3];        // [64]
    float*           out = (float*)d_out;               // [N, 64]

    const int N = in_sizes[0] / D_FEAT;
    const int E = in_sizes[1] / 2;
    const long long* src = ei;       // edge_index[0]
    const long long* dst = ei + E;   // edge_index[1]

    // workspace: xw [N*64 f32] | deg/dis [N f32]
    float* xw  = (float*)d_ws;
    float* deg = xw + (size_t)N * D_FEAT;

    const int total = N * D_FEAT;

    // 1) xw = x @ W  (N is a multiple of 16 here: 50000 = 3125*16)
    gcn_gemm_wmma<<<N / 16, 128, 0, stream>>>(x, W, xw);

    // 2) degree of (A + I) over targets, then dis = rsqrt(deg) in place
    k_deg_init <<<(N + 255) / 256, 256, 0, stream>>>(deg, N);
    k_deg_count<<<(E + 255) / 256, 256, 0, stream>>>(dst, deg, E);
    k_dis      <<<(N + 255) / 256, 256, 0, stream>>>(deg, N);

    // 3) out = b + dis^2 * xw   (self-loop term; full overwrite)
    k_out_init<<<(total + 255) / 256, 256, 0, stream>>>(xw, deg, b, out, total);

    // 4) edge messages: out[dst] += dis[src]*dis[dst] * xw[src]
    k_scatter<<<(E * 16 + 255) / 256, 256, 0, stream>>>(src, dst, xw, deg, out, E);

    // 5) ReLU
    k_relu<<<(total + 255) / 256, 256, 0, stream>>>(out, total);
}